// RGCNEncoder_6897717478007
// MI455X (gfx1250) — compile-verified
//
#include <hip/hip_runtime.h>

// ---------------------------------------------------------------------------
// RGCN (block-diagonal, mean aggregation), 2 layers, for MI455X / gfx1250.
// N=50000, d=256, R=16, E=800000, B=64 blocks of 4x4.
//
// Design:
//  * root GEMM  : fp32 WMMA V_WMMA_F32_16X16X4_F32. Each wave32 computes a
//                 64(M)x16(N) strip = 4 M-tiles sharing one B fragment, so
//                 Root is loaded once per 4 WMMAs (4x less L2 B-traffic).
//  * aggregation: per-edge block-diag transform + fp32 global atomics
//                 (E = N*R here, so per-edge costs the same flops as
//                  per-node-per-relation but needs no [N,R,d] buffer).
//  * workspace  : cnt[N][R] int (3.2MB) + A[N][d] f32 (51.2MB).
// ---------------------------------------------------------------------------

typedef __attribute__((ext_vector_type(2))) float v2f;
typedef __attribute__((ext_vector_type(8))) float v8f;

#define DIMD 256
#define NBLK 64
#define DB   4
#define MT   4      // M-tiles per wave in the GEMM

__device__ __forceinline__ void atomicAddF32(float* p, float v) {
    __hip_atomic_fetch_add(p, v, __ATOMIC_RELAXED, __HIP_MEMORY_SCOPE_AGENT);
}

__global__ void rgcn_zero_i32(int* __restrict__ p, int n) {
    int i = blockIdx.x * blockDim.x + threadIdx.x;
    if (i < n) p[i] = 0;
}

// cnt[dst][etype] += 1 over all edges
__global__ void rgcn_count_edges(const int* __restrict__ dst,
                                 const int* __restrict__ etype,
                                 int* __restrict__ cnt, int E, int R) {
    int e = blockIdx.x * blockDim.x + threadIdx.x;
    if (e < E) {
        __hip_atomic_fetch_add(&cnt[dst[e] * R + etype[e]], 1,
                               __ATOMIC_RELAXED, __HIP_MEMORY_SCOPE_AGENT);
    }
}

// Out[m][n] = sum_k X[m][k] * Root[k][n] + bias[n]
// One wave32 per 64x16 output strip (4 M-tiles x 1 N-tile); K-loop of
// 64 k-steps, each: 1 B fragment (2 scalar loads) feeding 4 fp32 WMMAs.
// A layout (16x4 f32): lanes 0-15 hold K=0,1 ; lanes 16-31 hold K=2,3 (M=lane&15).
// C/D layout: VGPR v -> M = v + (lane<16 ? 0 : 8), N = lane&15.
__global__ void rgcn_root_gemm_wmma(const float* __restrict__ X,
                                    const float* __restrict__ Root,
                                    const float* __restrict__ bias,
                                    float* __restrict__ Out, int Nn) {
    const int wave   = blockIdx.x * (blockDim.x >> 5) + (threadIdx.x >> 5);
    const int lane   = threadIdx.x & 31;
    const int tilesN = DIMD / 16;                       // 16
    const int tilesM = Nn / 16;                         // 3125 (N multiple of 16)
    const int groupsM = (tilesM + MT - 1) / MT;         // 782
    if (wave >= groupsM * tilesN) return;               // wave-uniform: EXEC all-1s
    const int gM    = wave / tilesN;
    const int tileN = wave % tilesN;

    const int l15  = lane & 15;
    const int koff = (lane < 16) ? 0 : 2;
    const int col  = tileN * 16 + l15;

    const float bv = bias[col];
    v8f acc[MT];
#pragma unroll
    for (int mt = 0; mt < MT; ++mt)
#pragma unroll
        for (int i = 0; i < 8; ++i) acc[mt][i] = bv;

    // Row bases, clamped (wave-uniform) so the tail group stays in-bounds;
    // duplicate tiles are computed but their stores are suppressed below.
    const float* xrow[MT];
#pragma unroll
    for (int mt = 0; mt < MT; ++mt) {
        int rb = (gM * MT + mt) * 16;
        if (rb > Nn - 16) rb = Nn - 16;
        xrow[mt] = X + (size_t)(rb + l15) * DIMD;
    }

#pragma unroll 4
    for (int k0 = 0; k0 < DIMD; k0 += 4) {
        v2f b;
        b.x = Root[(size_t)(k0 + koff) * DIMD + col];       // Root[k][n]
        b.y = Root[(size_t)(k0 + koff + 1) * DIMD + col];
#pragma unroll
        for (int mt = 0; mt < MT; ++mt) {
            v2f a = *(const v2f*)(xrow[mt] + k0 + koff);    // X[m][k], X[m][k+1]
            acc[mt] = __builtin_amdgcn_wmma_f32_16x16x4_f32(
                /*neg_a=*/false, a, /*neg_b=*/false, b,
                /*c_mod=*/(short)0, acc[mt], /*reuse_a=*/false, /*reuse_b=*/false);
        }
    }

#pragma unroll
    for (int mt = 0; mt < MT; ++mt) {
        const int tm = gM * MT + mt;
        if (tm < tilesM) {                                   // wave-uniform guard
            const int rbase = tm * 16 + ((lane < 16) ? 0 : 8);
#pragma unroll
            for (int v = 0; v < 8; ++v)
                Out[(size_t)(rbase + v) * DIMD + col] = acc[mt][v];
        }
    }
}

// Per-edge: Out[dst] += (blockdiag(W[etype]) . X[src]) / max(cnt[dst][etype],1)
// One wave per edge; lane handles blocks 2*lane, 2*lane+1 (8 contiguous outputs).
__global__ void rgcn_edge_scatter(const float* __restrict__ Xin,
                                  const int* __restrict__ src,
                                  const int* __restrict__ dst,
                                  const int* __restrict__ etype,
                                  const float* __restrict__ W,   // [R][64][4][4]
                                  const int* __restrict__ cnt,   // [N][R]
                                  float* __restrict__ Out, int E, int R) {
    const int wave = blockIdx.x * (blockDim.x >> 5) + (threadIdx.x >> 5);
    if (wave >= E) return;
    const int lane = threadIdx.x & 31;

    const int s = src[wave];
    const int t = dst[wave];
    const int r = etype[wave];
    const float invc = 1.0f / fmaxf((float)cnt[t * R + r], 1.0f);

    const float* xs = Xin + (size_t)s * DIMD;
    float*       od = Out + (size_t)t * DIMD;

#pragma unroll
    for (int bb = 0; bb < 2; ++bb) {
        const int b = lane * 2 + bb;
        const float4 h = *(const float4*)(xs + b * DB);
        const float* Wb = W + ((size_t)(r * NBLK + b)) * (DB * DB); // [i][o] row-major
#pragma unroll
        for (int o = 0; o < DB; ++o) {
            float acc = h.x * Wb[0 * DB + o] + h.y * Wb[1 * DB + o] +
                        h.z * Wb[2 * DB + o] + h.w * Wb[3 * DB + o];
            atomicAddF32(od + b * DB + o, acc * invc);
        }
    }
}

__global__ void rgcn_relu(float* __restrict__ p, int n) {
    int i = blockIdx.x * blockDim.x + threadIdx.x;
    if (i < n) p[i] = fmaxf(p[i], 0.0f);
}

extern "C" void kernel_launch(void* const* d_in, const int* in_sizes, int n_in,
                              void* d_out, int out_size, void* d_ws, size_t ws_size,
                              hipStream_t stream) {
    const float* node_emb   = (const float*)d_in[0];
    const int*   edge_index = (const int*)  d_in[1];
    const int*   edge_type  = (const int*)  d_in[2];
    const float* W1         = (const float*)d_in[3];
    const float* root1      = (const float*)d_in[4];
    const float* bias1      = (const float*)d_in[5];
    const float* W2         = (const float*)d_in[6];
    const float* root2      = (const float*)d_in[7];
    const float* bias2      = (const float*)d_in[8];
    float* out = (float*)d_out;

    const int N = in_sizes[0] / DIMD;                 // 50000
    const int E = in_sizes[2];                        // 800000
    const int R = in_sizes[3] / (NBLK * DB * DB);     // 16

    const int* srcp = edge_index;        // edge_index[0]
    const int* dstp = edge_index + E;    // edge_index[1]

    // workspace carve: cnt[N*R] int, then A[N*DIMD] float
    char* ws = (char*)d_ws;
    int* cnt = (int*)ws;
    size_t cntBytes = (((size_t)N * R * sizeof(int)) + 255) & ~(size_t)255;
    float* A = (float*)(ws + cntBytes);

    // ---- shared edge-count table (same graph for both layers) ----
    {
        int n = N * R;
        rgcn_zero_i32<<<(n + 255) / 256, 256, 0, stream>>>(cnt, n);
        rgcn_count_edges<<<(E + 255) / 256, 256, 0, stream>>>(dstp, edge_type, cnt, E, R);
    }

    const int tilesM      = N / 16;
    const int groupsM     = (tilesM + MT - 1) / MT;
    const int gemmWaves   = groupsM * (DIMD / 16);
    const int gemmThreads = 128;                      // 4 waves / block
    const int gemmBlocks  = (gemmWaves + 3) / 4;
    const int edgeThreads = 256;                      // 8 waves / block
    const int edgeBlocks  = (E + 7) / 8;
    const int reluBlocks  = (N * DIMD + 255) / 256;

    // ---- layer 1: A = relu(node_emb @ root1 + bias1 + scatter(W1)) ----
    rgcn_root_gemm_wmma<<<gemmBlocks, gemmThreads, 0, stream>>>(node_emb, root1, bias1, A, N);
    rgcn_edge_scatter<<<edgeBlocks, edgeThreads, 0, stream>>>(node_emb, srcp, dstp, edge_type,
                                                              W1, cnt, A, E, R);
    rgcn_relu<<<reluBlocks, 256, 0, stream>>>(A, N * DIMD);

    // ---- layer 2: out = A @ root2 + bias2 + scatter(W2) ----
    rgcn_root_gemm_wmma<<<gemmBlocks, gemmThreads, 0, stream>>>(A, root2, bias2, out, N);
    rgcn_edge_scatter<<<edgeBlocks, edgeThreads, 0, stream>>>(A, srcp, dstp, edge_type,
                                                              W2, cnt, out, E, R);
}